// Medical_Attention_1838246003405
// MI455X (gfx1250) — compile-verified
//
#include <hip/hip_runtime.h>
#include <hip/hip_bf16.h>
#include <math.h>

// Problem constants
#define AGENT_NUM 5
#define BB 16
#define TT 2048
#define DD 768

typedef __attribute__((ext_vector_type(2))) float v2f;
typedef __attribute__((ext_vector_type(8))) float v8f;

// ---------------------------------------------------------------------------
// K1a: partial masked column sums of q over T chunks.
// grid (B, 32), block 256. Each block reduces 64 t-rows for all 768 d.
// ---------------------------------------------------------------------------
__global__ __launch_bounds__(256) void k_partial_sum(const float* __restrict__ q,
                                                     const int* __restrict__ mask,
                                                     float* __restrict__ partial) {
  int b = blockIdx.x, tc = blockIdx.y;
  int tid = threadIdx.x;
  const float* qb = q + (size_t)b * TT * DD;
  const int* mb = mask + (size_t)b * TT;
  float a0 = 0.f, a1 = 0.f, a2 = 0.f;
  int d0 = tid, d1 = tid + 256, d2 = tid + 512;
  for (int j = 0; j < 64; ++j) {
    int t = tc * 64 + j;
    if (mb[t] != 0) {
      const float* r = qb + (size_t)t * DD;
      a0 += r[d0]; a1 += r[d1]; a2 += r[d2];
    }
  }
  float* p = partial + (size_t)(b * 32 + tc) * DD;
  p[d0] = a0; p[d1] = a1; p[d2] = a2;
}

// ---------------------------------------------------------------------------
// K1b: reduce the 32 partials -> sumq[B,D] (un-normalized mean, 1/T folded later)
// ---------------------------------------------------------------------------
__global__ __launch_bounds__(256) void k_reduce_sum(const float* __restrict__ partial,
                                                    float* __restrict__ sumq) {
  int idx = blockIdx.x * 256 + threadIdx.x;       // b*DD + d
  if (idx >= BB * DD) return;
  int b = idx / DD, d = idx - b * DD;
  float s = 0.f;
  for (int c = 0; c < 32; ++c) s += partial[(size_t)(b * 32 + c) * DD + d];
  sumq[idx] = s;
}

// ---------------------------------------------------------------------------
// K2: per-(b,t) dots: u = q.mq, s = k.mq  (one wave32 per t, shuffle reduce)
// grid B*T/8, block 256 (8 waves)
// ---------------------------------------------------------------------------
__global__ __launch_bounds__(256) void k_scores(const float* __restrict__ q,
                                                const float* __restrict__ k,
                                                const float* __restrict__ sumq,
                                                float* __restrict__ u,
                                                float* __restrict__ s) {
  int gid = blockIdx.x * 8 + (threadIdx.x >> 5);  // wave id = b*TT + t
  int lane = threadIdx.x & 31;
  int b = gid >> 11;                               // / TT
  const float* qr = q + (size_t)gid * DD;
  const float* kr = k + (size_t)gid * DD;
  const float* mq = sumq + (size_t)b * DD;
  float au = 0.f, as = 0.f;
  for (int d = lane; d < DD; d += 32) {
    float m = mq[d];
    au += qr[d] * m;
    as += kr[d] * m;
  }
  for (int off = 16; off; off >>= 1) {
    au += __shfl_xor(au, off, 32);
    as += __shfl_xor(as, off, 32);
  }
  if (lane == 0) {
    const float invT = 1.0f / (float)TT;
    u[gid] = au * invT;
    s[gid] = as * invT;
  }
}

// ---------------------------------------------------------------------------
// K3: stage-1 softmax over T for each (b, a); rows a>=5 zero-padded.
// grid (B,16), block 256, 8 elements/thread.
// ---------------------------------------------------------------------------
__global__ __launch_bounds__(256) void k_softmax1(const float* __restrict__ s,
                                                  const float* __restrict__ b1,
                                                  float* __restrict__ attn) {
  int b = blockIdx.x, a = blockIdx.y;
  int tid = threadIdx.x;
  float* out = attn + (size_t)(b * 16 + a) * TT;
  if (a >= AGENT_NUM) {
    for (int j = 0; j < 8; ++j) out[tid + 256 * j] = 0.f;
    return;
  }
  __shared__ float red[256];
  const float* sb = s + (size_t)b * TT;
  const float* bb = b1 + (size_t)a * TT;
  float vals[8];
  float m = -INFINITY;
  for (int j = 0; j < 8; ++j) {
    int t = tid + 256 * j;
    float x = sb[t] + bb[t];
    vals[j] = x;
    m = fmaxf(m, x);
  }
  red[tid] = m; __syncthreads();
  for (int off = 128; off; off >>= 1) {
    if (tid < off) red[tid] = fmaxf(red[tid], red[tid + off]);
    __syncthreads();
  }
  m = red[0]; __syncthreads();
  float sum = 0.f;
  for (int j = 0; j < 8; ++j) { vals[j] = expf(vals[j] - m); sum += vals[j]; }
  red[tid] = sum; __syncthreads();
  for (int off = 128; off; off >>= 1) {
    if (tid < off) red[tid] += red[tid + off];
    __syncthreads();
  }
  float inv = 1.0f / red[0];
  for (int j = 0; j < 8; ++j) out[tid + 256 * j] = vals[j] * inv;
}

// ---------------------------------------------------------------------------
// K3b: stage-2 softmax over A=5 per (b,t); write padded-to-8 weights.
// grid B*T/256, block 256.
// ---------------------------------------------------------------------------
__global__ __launch_bounds__(256) void k_softmax2(const float* __restrict__ u,
                                                  const float* __restrict__ b2,
                                                  float* __restrict__ wpad) {
  int idx = blockIdx.x * 256 + threadIdx.x;   // b*TT + t
  int t = idx & (TT - 1);
  float uu = u[idx];
  float v[AGENT_NUM];
  float m = -INFINITY;
  for (int a = 0; a < AGENT_NUM; ++a) {
    v[a] = uu + b2[t * AGENT_NUM + a];
    m = fmaxf(m, v[a]);
  }
  float sum = 0.f;
  for (int a = 0; a < AGENT_NUM; ++a) { v[a] = expf(v[a] - m); sum += v[a]; }
  float inv = 1.0f / sum;
  float* o = wpad + (size_t)idx * 8;
  for (int a = 0; a < AGENT_NUM; ++a) o[a] = v[a] * inv;
  o[5] = 0.f; o[6] = 0.f; o[7] = 0.f;
}

// ---------------------------------------------------------------------------
// K4: agent_v = attn_pad[16 x T] @ v[T x D] per batch via V_WMMA_F32_16X16X4_F32.
// grid (B, 48), block 256 (8 waves). Wave w accumulates its 256-row T-chunk,
// then the 8 partial C tiles are summed serially (w = 0..7) through LDS (fixed order).
// Output padded to 8 rows (rows 5..7 = 0).
// A layout: lane l holds row l%16, K pair base 2*(l>>4).
// B layout: lane l holds col l%16, K rows 2*(l>>4)+{0,1}.
// C layout: vgpr r, lanes 0-15 -> M=r, lanes 16-31 -> M=r+8.
// ---------------------------------------------------------------------------
__global__ __launch_bounds__(256) void k_agent_v(const float* __restrict__ v,
                                                 const float* __restrict__ attn,
                                                 float* __restrict__ avpad) {
  int b = blockIdx.x;
  int nt = blockIdx.y;                    // d tile 0..47
  int wave = threadIdx.x >> 5;
  int lane = threadIdx.x & 31;
  int row = lane & 15;
  int half = lane >> 4;
  int col = nt * 16 + row;
  const float* vb = v + (size_t)b * TT * DD;
  const float* ab = attn + (size_t)b * 16 * TT;
  v8f c = {0.f, 0.f, 0.f, 0.f, 0.f, 0.f, 0.f, 0.f};
  int t0 = wave * (TT / 8);               // 256-row chunk per wave
  for (int t = t0; t < t0 + TT / 8; t += 4) {
    int ka = t + 2 * half;
    v2f afrag, bfrag;
    afrag.x = ab[(size_t)row * TT + ka];
    afrag.y = ab[(size_t)row * TT + ka + 1];
    bfrag.x = vb[(size_t)ka * DD + col];
    bfrag.y = vb[(size_t)(ka + 1) * DD + col];
    c = __builtin_amdgcn_wmma_f32_16x16x4_f32(false, afrag, false, bfrag,
                                              (short)0, c, false, false);
  }
  // Cross-wave reduction of the 8 partial C tiles. Only rows 0..7 are nonzero
  // (A rows 5..15 are zero-padded), and those live in lanes 0..15, vgprs 0..7.
  __shared__ float red[8][8][16];         // [wave][row r][col n]
  if (lane < 16) {
    for (int r = 0; r < 8; ++r) red[wave][r][lane] = c[r];
  }
  __syncthreads();
  int tid = threadIdx.x;
  if (tid < 128) {                        // one thread per (r, n)
    int r = tid >> 4, n = tid & 15;
    float s = 0.f;
    for (int w = 0; w < 8; ++w) s += red[w][r][n];
    avpad[((size_t)b * 8 + r) * DD + nt * 16 + n] = s;
  }
}

// ---------------------------------------------------------------------------
// K5: x = wpad[T x 8] @ avpad[8 x D] per batch via 2x V_WMMA_F32_16X16X4_F32.
// grid (B, T/16), block 256 (8 waves); wave w covers d-tiles w, w+8, ...
// ---------------------------------------------------------------------------
__global__ __launch_bounds__(256) void k_output(const float* __restrict__ wpad,
                                                const float* __restrict__ avpad,
                                                float* __restrict__ x) {
  int b = blockIdx.x;
  int mt = blockIdx.y;                    // t tile 0..127
  int wave = threadIdx.x >> 5;
  int lane = threadIdx.x & 31;
  int row = lane & 15;
  int half = lane >> 4;
  const float* wb = wpad + ((size_t)b * TT + (size_t)mt * 16) * 8;
  const float* av = avpad + (size_t)b * 8 * DD;
  // A fragments (shared by all d-tiles of this wave)
  v2f a0, a1;
  {
    int k0 = 2 * half;
    int k1 = 4 + 2 * half;
    a0.x = wb[row * 8 + k0]; a0.y = wb[row * 8 + k0 + 1];
    a1.x = wb[row * 8 + k1]; a1.y = wb[row * 8 + k1 + 1];
  }
  float* xo = x + ((size_t)b * TT + (size_t)mt * 16) * DD;
  for (int nt = wave; nt < 48; nt += 8) {
    int col = nt * 16 + row;
    v2f b0, b1;
    b0.x = av[(size_t)(2 * half) * DD + col];
    b0.y = av[(size_t)(2 * half + 1) * DD + col];
    b1.x = av[(size_t)(4 + 2 * half) * DD + col];
    b1.y = av[(size_t)(4 + 2 * half + 1) * DD + col];
    v8f c = {0.f, 0.f, 0.f, 0.f, 0.f, 0.f, 0.f, 0.f};
    c = __builtin_amdgcn_wmma_f32_16x16x4_f32(false, a0, false, b0,
                                              (short)0, c, false, false);
    c = __builtin_amdgcn_wmma_f32_16x16x4_f32(false, a1, false, b1,
                                              (short)0, c, false, false);
    for (int r = 0; r < 8; ++r)
      xo[(size_t)(r + 8 * half) * DD + col] = c[r];
  }
}

// ---------------------------------------------------------------------------
// Launcher
// ---------------------------------------------------------------------------
extern "C" void kernel_launch(void* const* d_in, const int* in_sizes, int n_in,
                              void* d_out, int out_size, void* d_ws, size_t ws_size,
                              hipStream_t stream) {
  const float* qkv = (const float*)d_in[0];           // [3,B,T,D]
  const int* mask  = (const int*)d_in[1];             // [B,T]
  const float* b1  = (const float*)d_in[2];           // [A,T]
  const float* b2  = (const float*)d_in[3];           // [T,A]
  float* x = (float*)d_out;                           // [B,T,D]

  const float* q = qkv;
  const float* k = qkv + (size_t)BB * TT * DD;
  const float* v = qkv + 2 * (size_t)BB * TT * DD;

  float* ws = (float*)d_ws;
  float* partial = ws;                                // B*32*D = 393216
  float* sumq    = partial + (size_t)BB * 32 * DD;    // B*D    =  12288
  float* u       = sumq + (size_t)BB * DD;            // B*T    =  32768
  float* s       = u + (size_t)BB * TT;               // B*T    =  32768
  float* attn    = s + (size_t)BB * TT;               // B*16*T = 524288
  float* wpad    = attn + (size_t)BB * 16 * TT;       // B*T*8  = 262144
  float* avpad   = wpad + (size_t)BB * TT * 8;        // B*8*D  =  98304

  // K1: masked column sums of q
  k_partial_sum<<<dim3(BB, 32), 256, 0, stream>>>(q, mask, partial);
  k_reduce_sum<<<(BB * DD + 255) / 256, 256, 0, stream>>>(partial, sumq);

  // K2: u[b,t] = q.mq, s[b,t] = k.mq (1/T folded)
  k_scores<<<(BB * TT) / 8, 256, 0, stream>>>(q, k, sumq, u, s);

  // K3: softmaxes
  k_softmax1<<<dim3(BB, 16), 256, 0, stream>>>(s, b1, attn);
  k_softmax2<<<(BB * TT) / 256, 256, 0, stream>>>(u, b2, wpad);

  // K4: agent_v via WMMA (reads v once, 8 waves/tile for latency hiding)
  k_agent_v<<<dim3(BB, DD / 16), 256, 0, stream>>>(v, attn, avpad);

  // K5: x via WMMA (writes output once)
  k_output<<<dim3(BB, TT / 16), 256, 0, stream>>>(wpad, avpad, x);
}